// Explore_decoder_add_43636867727569
// MI455X (gfx1250) — compile-verified
//
#include <hip/hip_runtime.h>
#include <math.h>

#define NB 16
#define NS 200
#define ND 128
#define NVOCAB 100000

typedef float v2f __attribute__((ext_vector_type(2)));
typedef float v8f __attribute__((ext_vector_type(8)));

// ---------------------------------------------------------------------------
// Kernel 1: attention head -> concat([x0, c_s]) (16 x 256) into workspace
// one block per batch, 256 threads (8 waves)
// ---------------------------------------------------------------------------
__global__ __launch_bounds__(256) void attn_kernel(
    const float* __restrict__ x,  const float* __restrict__ Wq,
    const float* __restrict__ bq, const float* __restrict__ Wk,
    const float* __restrict__ bk, const float* __restrict__ Wv,
    const float* __restrict__ bv, float* __restrict__ cc)
{
    __shared__ float s_x0[ND];
    __shared__ float s_k[ND];
    __shared__ float s_sc[NS];
    __shared__ float s_red[256];

    const int b = blockIdx.x;
    const int t = threadIdx.x;
    const float* xb = x + (size_t)b * NS * ND;

    if (t < ND) s_x0[t] = xb[t];
    __syncthreads();

    // k = x0 @ Wk + bk   (one output per thread t<128)
    if (t < ND) {
        float acc = bk[t];
        for (int j = 0; j < ND; ++j) acc += s_x0[j] * Wk[j * ND + t];
        s_k[t] = acc;
    }
    __syncthreads();

    const int wave = t >> 5;
    const int lane = t & 31;
    const float bv0 = bv[0];

    // scores[s] = (tanh(x_s@Wq + bq + k)) . Wv + bv ; one s per wave per iter
    for (int s = wave; s < NS; s += 8) {
        const float* xs = xb + s * ND;
        float partial = 0.f;
        #pragma unroll
        for (int dd = 0; dd < 4; ++dd) {
            const int d = lane + dd * 32;
            float acc = bq[d];
            for (int j = 0; j < ND; ++j) acc += xs[j] * Wq[j * ND + d];
            partial += tanhf(acc + s_k[d]) * Wv[d];
        }
        for (int off = 16; off > 0; off >>= 1)
            partial += __shfl_xor(partial, off, 32);
        if (lane == 0) s_sc[s] = partial + bv0;
    }
    __syncthreads();

    // softmax over the 200 scores
    s_red[t] = (t < NS) ? s_sc[t] : -__builtin_inff();
    __syncthreads();
    for (int off = 128; off > 0; off >>= 1) {
        if (t < off) s_red[t] = fmaxf(s_red[t], s_red[t + off]);
        __syncthreads();
    }
    const float mx = s_red[0];
    __syncthreads();

    float e = 0.f;
    if (t < NS) { e = __expf(s_sc[t] - mx); s_sc[t] = e; }
    s_red[t] = e;
    __syncthreads();
    for (int off = 128; off > 0; off >>= 1) {
        if (t < off) s_red[t] += s_red[t + off];
        __syncthreads();
    }
    const float inv = 1.f / s_red[0];
    __syncthreads();

    // c_s[d] = sum_s p[s] * x[b,s,d]; emit concat([x0, c_s])
    if (t < ND) {
        float acc = 0.f;
        for (int s = 0; s < NS; ++s) acc += xb[s * ND + t] * s_sc[s];
        cc[b * 2 * ND + t]      = s_x0[t];
        cc[b * 2 * ND + ND + t] = acc * inv;
    }
}

// ---------------------------------------------------------------------------
// Kernel 2: logits = cc(16x256) @ Wec(256x100000) + bec  via V_WMMA_F32_16X16X4_F32
// 8 waves / block, one 16x16 output tile per wave; 6250 tiles total.
// ---------------------------------------------------------------------------
__global__ __launch_bounds__(256) void gemm_kernel(
    const float* __restrict__ cc, const float* __restrict__ Wec,
    const float* __restrict__ bec, float* __restrict__ out)
{
    __shared__ float sA[16 * 256];   // the whole A matrix (16 KB)
    const int t = threadIdx.x;
    for (int i = t; i < 16 * 256; i += 256) sA[i] = cc[i];
    __syncthreads();

    const int wave = t >> 5;
    const int lane = t & 31;
    const int tile = blockIdx.x * 8 + wave;
    if (tile >= NVOCAB / 16) return;          // wave-uniform exit (after barrier)

    const int n0 = tile * 16;
    const int m  = lane & 15;                 // A row / D column index
    const int kc = (lane >> 4) << 1;          // K sub-offset: 0 (lanes 0-15) or 2
    const int n  = n0 + m;                    // this lane's output column

    // fold bias into the accumulator: every D element of this lane has column n
    v8f acc;
    const float bias = bec[n];
    #pragma unroll
    for (int i = 0; i < 8; ++i) acc[i] = bias;

    const float* __restrict__ wp = Wec + n;
    #pragma unroll 4
    for (int kb = 0; kb < 64; ++kb) {
        const int kk = kb * 4 + kc;
        v2f a = *(const v2f*)&sA[m * 256 + kk];          // A[m][kk], A[m][kk+1]
        v2f bfrag;
        bfrag.x = wp[(size_t)kk * NVOCAB];               // B[kk][n]
        bfrag.y = wp[(size_t)(kk + 1) * NVOCAB];         // B[kk+1][n]
        acc = __builtin_amdgcn_wmma_f32_16x16x4_f32(
            false, a, false, bfrag, (short)0, acc, false, false);
    }

    const int rbase = (lane >> 4) * 8;        // D rows: i + 8*(lane>=16)
    #pragma unroll
    for (int i = 0; i < 8; ++i)
        out[(size_t)(rbase + i) * NVOCAB + n] = acc[i];
}

// ---------------------------------------------------------------------------
// Kernel 3: scatter -inf at seen ids (ids 0/1 excluded)
// ---------------------------------------------------------------------------
__global__ void mask_kernel(const int* __restrict__ ids, float* __restrict__ out)
{
    const int i = blockIdx.x * blockDim.x + threadIdx.x;
    if (i >= NB * NS) return;
    const int b  = i / NS;
    const int id = ids[i];
    if (id != 0 && id != 1)
        out[(size_t)b * NVOCAB + id] = -__builtin_inff();
}

// ---------------------------------------------------------------------------
// Kernel 4: softmax over 100000 per row; one block per row (L2-resident data)
// ---------------------------------------------------------------------------
__global__ __launch_bounds__(1024) void softmax_kernel(float* __restrict__ out)
{
    __shared__ float red[1024];
    const int b = blockIdx.x;
    const int t = threadIdx.x;
    float* row = out + (size_t)b * NVOCAB;

    float m = -__builtin_inff();
    for (int i = t; i < NVOCAB; i += 1024) m = fmaxf(m, row[i]);
    red[t] = m;
    __syncthreads();
    for (int off = 512; off > 0; off >>= 1) {
        if (t < off) red[t] = fmaxf(red[t], red[t + off]);
        __syncthreads();
    }
    m = red[0];
    __syncthreads();

    float s = 0.f;
    for (int i = t; i < NVOCAB; i += 1024) {
        const float e = __expf(row[i] - m);   // exp(-inf - m) == 0
        row[i] = e;
        s += e;
    }
    red[t] = s;
    __syncthreads();
    for (int off = 512; off > 0; off >>= 1) {
        if (t < off) red[t] += red[t + off];
        __syncthreads();
    }
    const float inv = 1.f / red[0];
    __syncthreads();

    for (int i = t; i < NVOCAB; i += 1024) row[i] *= inv;
}

extern "C" void kernel_launch(void* const* d_in, const int* in_sizes, int n_in,
                              void* d_out, int out_size, void* d_ws, size_t ws_size,
                              hipStream_t stream)
{
    const float* x   = (const float*)d_in[0];
    const int*   ids = (const int*)  d_in[1];
    const float* Wq  = (const float*)d_in[2];
    const float* bq  = (const float*)d_in[3];
    const float* Wk  = (const float*)d_in[4];
    const float* bk  = (const float*)d_in[5];
    const float* Wv  = (const float*)d_in[6];
    const float* bv  = (const float*)d_in[7];
    const float* Wec = (const float*)d_in[8];
    const float* bec = (const float*)d_in[9];
    float* out = (float*)d_out;
    float* cc  = (float*)d_ws;                 // 16*256 floats of scratch

    attn_kernel<<<NB, 256, 0, stream>>>(x, Wq, bq, Wk, bk, Wv, bv, cc);
    gemm_kernel<<<(NVOCAB / 16 + 7) / 8, 256, 0, stream>>>(cc, Wec, bec, out);
    mask_kernel<<<(NB * NS + 255) / 256, 256, 0, stream>>>(ids, out);
    softmax_kernel<<<NB, 1024, 0, stream>>>(out);
}